// CustomLoss_37160057045144
// MI455X (gfx1250) — compile-verified
//
#include <hip/hip_runtime.h>
#include <hip/hip_bf16.h>
#include <math.h>

// Problem shape (fixed by reference setup_inputs): B=256, N=2048, M=512.
#define BB  256
#define NN  2048
#define MM  512
#define TPB 512   // 16 wave32s per workgroup

#ifndef __has_builtin
#define __has_builtin(x) 0
#endif

#if __has_builtin(__builtin_amdgcn_global_load_async_to_lds_b32)
#define USE_ASYNC_LDS 1
#else
#define USE_ASYNC_LDS 0
#endif

#if __has_builtin(__builtin_amdgcn_wmma_f32_16x16x4_f32)
#define USE_WMMA_RED 1
typedef __attribute__((ext_vector_type(2))) float v2f;
typedef __attribute__((ext_vector_type(8))) float v8f;
#else
#define USE_WMMA_RED 0
#endif

// generic -> AS1 / AS3 via integer detour (ISA 10.2: LDS generic address low
// 32 bits == LDS offset; global generic == AS1). Builtin takes typed int*.
#if USE_ASYNC_LDS
typedef __attribute__((address_space(1))) int as1_int;
typedef __attribute__((address_space(3))) int as3_int;
__device__ __forceinline__ as1_int* to_glob(const void* p) {
  return (as1_int*)(unsigned long long)(uintptr_t)p;
}
__device__ __forceinline__ as3_int* to_lds(const void* p) {
  return (as3_int*)(unsigned int)(uintptr_t)p;
}
#endif

__device__ __forceinline__ void async_wait0() {
#if USE_ASYNC_LDS
#if __has_builtin(__builtin_amdgcn_s_wait_asynccnt)
  __builtin_amdgcn_s_wait_asynccnt(0);
#else
  asm volatile("s_wait_asynccnt 0" ::: "memory");
#endif
#endif
}

// ---------------------------------------------------------------------------
// Phase 1: one workgroup per batch element. Produces 5 partial loss sums.
// ---------------------------------------------------------------------------
__global__ __launch_bounds__(TPB) void loss_build_kernel(
    const float* __restrict__ preds,     // [B, N, 5]  (p, x, y, a, b)
    const float* __restrict__ targets,   // [B, M, 5]  (valid, x, y, a, b)
    float* __restrict__ ws) {            // [B, 5] partial sums
  const int b    = blockIdx.x;
  const int tid  = threadIdx.x;
  const int lane = tid & 31;
  const int wid  = tid >> 5;

  __shared__ float                sT[MM * 5];  // targets, AoS (stride 5 -> conflict-free)
  __shared__ float                sP[MM * 5];  // top-512 confidence-sorted pred rows
  __shared__ alignas(16) unsigned sKey[NN];    // ~bits(conf): ascending == conf descending
  __shared__ float                sW[16 * 5];  // per-wave partial sums
  __shared__ int                  sNpos, sNtarg;

  const float* predB = preds   + (size_t)b * NN * 5;
  const float* targB = targets + (size_t)b * MM * 5;

  if (tid == 0) { sNpos = 0; sNtarg = 0; }
  __syncthreads();

  // ---- stage targets into LDS via CDNA5 async global->LDS path ----
#if USE_ASYNC_LDS
  #pragma unroll
  for (int j = 0; j < 5; ++j) {
    int i = tid + j * TPB;  // 0..2559, coalesced
    __builtin_amdgcn_global_load_async_to_lds_b32(to_glob(targB + i), to_lds(&sT[i]), 0, 0);
  }
#else
  #pragma unroll
  for (int j = 0; j < 5; ++j) { int i = tid + j * TPB; sT[i] = targB[i]; }
#endif

  // ---- load confidences, build sort keys, base BCE (all preds at t=0) ----
  float    pconf[4];
  unsigned mykey[4];
  int      myrank[4];
  int   nposLocal = 0;
  float accP = 0.f;   // BCE sum accumulator
  #pragma unroll
  for (int j = 0; j < 4; ++j) {
    int n = tid + j * TPB;
    float p = predB[(size_t)n * 5];
    pconf[j]  = p;
    unsigned k = ~__float_as_uint(p);   // conf in [0,1): bits monotone; ~ flips order
    mykey[j]  = k;
    sKey[n]   = k;
    nposLocal += (p > 0.5f) ? 1 : 0;
    accP += -fmaxf(log1pf(-p), -100.0f);   // t=0 term: -clip(log1p(-p), -100)
    myrank[j] = 0;
  }
  if (nposLocal) atomicAdd(&sNpos, nposLocal);

  async_wait0();
  __syncthreads();   // sKey + sT visible

  atomicAdd(&sNtarg, (sT[tid * 5] == 1.0f) ? 1 : 0);

  // ---- stable rank by counting: rank = #{j : key_j < key_n  or (==, j<n)} ----
  // b128 LDS reads: 4 keys per ds_load, broadcast across the wave.
  {
    const uint4* sKey4 = (const uint4*)sKey;
    for (int jj4 = 0; jj4 < NN / 4; ++jj4) {
      uint4 kv = sKey4[jj4];
      int jb = jj4 * 4;
      #pragma unroll
      for (int q = 0; q < 4; ++q) {
        int n = tid + q * TPB;
        unsigned mk = mykey[q];
        myrank[q] += (kv.x < mk || (kv.x == mk && (jb + 0) < n)) ? 1 : 0;
        myrank[q] += (kv.y < mk || (kv.y == mk && (jb + 1) < n)) ? 1 : 0;
        myrank[q] += (kv.z < mk || (kv.z == mk && (jb + 2) < n)) ? 1 : 0;
        myrank[q] += (kv.w < mk || (kv.w == mk && (jb + 3) < n)) ? 1 : 0;
      }
    }
  }
  // scatter top-512 sorted pred rows into LDS (ranks form a permutation -> no races)
  #pragma unroll
  for (int q = 0; q < 4; ++q) {
    int r = myrank[q];
    if (r < MM) {
      int n = tid + q * TPB;
      const float* row = predB + (size_t)n * 5;
      sP[r * 5 + 0] = pconf[q];
      sP[r * 5 + 1] = row[1];
      sP[r * 5 + 2] = row[2];
      sP[r * 5 + 3] = row[3];
      sP[r * 5 + 4] = row[4];
    }
  }
  __syncthreads();

  const int ntarg = sNtarg;
  const int npos  = sNpos;
  const int K     = (npos < ntarg) ? npos : ntarg;

  // ---- BCE correction: ranks < ntarg have t=1 (matched or row_first) ----
  if (tid < ntarg) {
    float p   = sP[tid * 5];
    float lp  = fmaxf(logf(p),    -100.0f);
    float l1m = fmaxf(log1pf(-p), -100.0f);
    accP += l1m - lp;   // replace -log1mp with -logp
  }

  float accX = 0.f, accY = 0.f, accA = 0.f, accB2 = 0.f;

  // ---- sequential greedy matching: wave 0 only ----
  // Lane owns targets m = k*32 + lane; (x,y) cached in VGPRs, 'used' is a
  // 16-bit register bitmask, so each serial step is pure VALU + 5 permute hops.
  if (wid == 0) {
    unsigned validmask = 0;
    float txr[16], tyr[16];
    #pragma unroll
    for (int k = 0; k < 16; ++k) {
      int m = k * 32 + lane;
      txr[k] = sT[m * 5 + 1];
      tyr[k] = sT[m * 5 + 2];
      if (sT[m * 5] == 1.0f) validmask |= (1u << k);
    }

    unsigned used = 0;
    for (int r = 0; r < K; ++r) {
      float px = sP[r * 5 + 1], py = sP[r * 5 + 2];
      float bestd = 3.4e38f;
      int   bestm = 0x7FFFFFFF;
      unsigned avail = validmask & ~used;
      #pragma unroll
      for (int k = 0; k < 16; ++k) {
        if (avail & (1u << k)) {
          float dx = txr[k] - px;
          float dy = tyr[k] - py;
          float d  = dx * dx + dy * dy;             // exact reference formula
          if (d < bestd) { bestd = d; bestm = k * 32 + lane; }  // strict <: lowest m on tie
        }
      }
      // wave32 argmin reduction (d, then lowest index)
      #pragma unroll
      for (int off = 16; off > 0; off >>= 1) {
        float od = __shfl_xor(bestd, off, 32);
        int   om = __shfl_xor(bestm, off, 32);
        if (od < bestd || (od == bestd && om < bestm)) { bestd = od; bestm = om; }
      }
      if ((bestm & 31) == lane) used |= (1u << (bestm >> 5));  // consume target
      if (lane == (r & 31)) {    // rotate accumulation across lanes
        float dx = sP[r * 5 + 1] - sT[bestm * 5 + 1];
        float dy = sP[r * 5 + 2] - sT[bestm * 5 + 2];
        float da = sP[r * 5 + 3] - sT[bestm * 5 + 3];
        float db = sP[r * 5 + 4] - sT[bestm * 5 + 4];
        accX += dx * dx; accY += dy * dy; accA += da * da; accB2 += db * db;
      }
    }

    // ranks K..ntarg-1 all get the (frozen) first unused valid target
    if (ntarg > K) {
      unsigned avail = validmask & ~used;
      int myf = avail ? ((__ffs(avail) - 1) * 32 + lane) : 0x7FFFFFFF;
      #pragma unroll
      for (int off = 16; off > 0; off >>= 1) {
        int om = __shfl_xor(myf, off, 32);
        myf = (om < myf) ? om : myf;
      }
      float fx = sT[myf * 5 + 1], fy = sT[myf * 5 + 2];
      float fa = sT[myf * 5 + 3], fb = sT[myf * 5 + 4];
      for (int r = K + lane; r < ntarg; r += 32) {
        float dx = sP[r * 5 + 1] - fx, dy = sP[r * 5 + 2] - fy;
        float da = sP[r * 5 + 3] - fa, db = sP[r * 5 + 4] - fb;
        accX += dx * dx; accY += dy * dy; accA += da * da; accB2 += db * db;
      }
    }
  }

  // ---- reduction of the 5 partial sums ----
  // Stage 1: per-wave shuffle reduction (no barriers).
  float accs[5] = {accX, accY, accA, accB2, accP};
  #pragma unroll
  for (int c = 0; c < 5; ++c) {
    #pragma unroll
    for (int off = 16; off > 0; off >>= 1) accs[c] += __shfl_xor(accs[c], off, 32);
  }
  if (lane == 0) {
    #pragma unroll
    for (int c = 0; c < 5; ++c) sW[wid * 5 + c] = accs[c];
  }
  __syncthreads();

  // Stage 2: fold the 16 wave partials on the matrix engine.
  // B = all-ones makes D[i][j] = sum_k A[i][k] independent of B's striping;
  // channels sit in A rows 0..4, 4 chained WMMAs consume 4 waves per K-slice.
#if USE_WMMA_RED
  if (wid == 0) {
    const int Mrow  = lane & 15;            // A-matrix row owned by this lane
    const int kbase = (lane < 16) ? 0 : 2;  // A rows: lanes 0-15 hold K=0,1; 16-31 hold K=2,3
    v2f bones = {1.0f, 1.0f};
    v8f d = {};
    #pragma unroll
    for (int s = 0; s < 4; ++s) {
      v2f a;
      a.x = (Mrow < 5) ? sW[(4 * s + kbase + 0) * 5 + Mrow] : 0.0f;
      a.y = (Mrow < 5) ? sW[(4 * s + kbase + 1) * 5 + Mrow] : 0.0f;
      d = __builtin_amdgcn_wmma_f32_16x16x4_f32(false, a, false, bones,
                                                (short)0, d, false, false);
    }
    // D element c in lane 0 == D[M=c][N=0] == total sum for channel c
    if (lane == 0) {
      #pragma unroll
      for (int c = 0; c < 5; ++c) ws[b * 5 + c] = d[c];
    }
  }
#else
  if (wid == 0 && lane == 0) {
    #pragma unroll
    for (int c = 0; c < 5; ++c) {
      float s = 0.f;
      for (int w = 0; w < 16; ++w) s += sW[w * 5 + c];
      ws[b * 5 + c] = s;
    }
  }
#endif
}

// ---------------------------------------------------------------------------
// Phase 2: reduce [B,5] partials -> 5 means (weights are all 1.0).
// Output order: [lossx, lossy, lossa, lossb, lossprob].
// ---------------------------------------------------------------------------
__global__ __launch_bounds__(BB) void loss_finish_kernel(
    const float* __restrict__ ws, float* __restrict__ out) {
  __shared__ float sRed[BB];
  const int t = threadIdx.x;
  const float inv = 1.0f / ((float)BB * (float)NN);
  for (int c = 0; c < 5; ++c) {
    __syncthreads();
    sRed[t] = ws[t * 5 + c];
    __syncthreads();
    for (int s = BB / 2; s > 0; s >>= 1) {
      if (t < s) sRed[t] += sRed[t + s];
      __syncthreads();
    }
    if (t == 0) out[c] = sRed[0] * inv;
  }
}

extern "C" void kernel_launch(void* const* d_in, const int* in_sizes, int n_in,
                              void* d_out, int out_size, void* d_ws, size_t ws_size,
                              hipStream_t stream) {
  const float* preds   = (const float*)d_in[0];   // [256,2048,5] f32
  const float* targets = (const float*)d_in[1];   // [256,512,5]  f32
  float* ws  = (float*)d_ws;                      // needs 256*5 floats
  float* out = (float*)d_out;                     // 5 floats

  loss_build_kernel<<<dim3(BB), dim3(TPB), 0, stream>>>(preds, targets, ws);
  loss_finish_kernel<<<dim3(1), dim3(BB), 0, stream>>>(ws, out);
}